// Processor_4252017623216
// MI455X (gfx1250) — compile-verified
//
#include <hip/hip_runtime.h>

#define N_LAYERS 10
#define DIM      128
#define N_NODES  30000
#define N_EDGES  240000
#define LN_EPS   1e-5f

typedef __attribute__((ext_vector_type(2))) float v2f;
typedef __attribute__((ext_vector_type(8))) float v8f;

#define MT 32                 // rows (edges/nodes) per workgroup
#define KE 384                // edge-MLP input K  (edges | recv | send)
#define KN 256                // node-MLP input K  (agg | nodes)
#define AE_STRIDE (KE + 4)    // 388 floats: 4-bank rotation/row, 16B aligned
#define AN_STRIDE (KN + 4)    // 260 floats
#define HS        (DIM + 4)   // 132 floats

__device__ __forceinline__ v8f wmma4(v2f a, v2f b, v8f c) {
  // D = A(16x4,f32) * B(4x16,f32) + C(16x16,f32)
  return __builtin_amdgcn_wmma_f32_16x16x4_f32(false, a, false, b, (short)0, c,
                                               false, false);
}

// One GEMM stage for a 32xK tile: out(32x128) = act(A(32xK) @ Wp(Kx128) + bias)
// Wp is K-pair packed: float2 at [(k>>1)*128 + col] = {W[k][col], W[k+1][col]}.
// Wave 'wave' owns output columns [32*wave, 32*wave+32); 2 M x 2 N subtiles.
template <int K, int AS>
__device__ __forceinline__ void gemm_stage(const float* __restrict__ As,
                                           const float2* __restrict__ Wp,
                                           const float* __restrict__ bias,
                                           float* __restrict__ dstH,
                                           int wave, int lane, bool relu) {
  v8f c00 = {}, c01 = {}, c10 = {}, c11 = {};
  const int m    = lane & 15;       // A row / B column within subtile
  const int kh   = lane >> 4;       // which K-pair of the quad this lane covers
  const int col0 = wave * 32 + m;   // first output column for this lane

#pragma unroll 4
  for (int k0 = 0; k0 < K; k0 += 4) {
    const int kb = k0 + 2 * kh;                 // A: columns kb, kb+1
    v2f a0, a1, b0, b1;
    const float* Ar0 = As + m * AS + kb;        // M-subtile 0 (rows 0..15)
    const float* Ar1 = Ar0 + 16 * AS;           // M-subtile 1 (rows 16..31)
    a0.x = Ar0[0]; a0.y = Ar0[1];
    a1.x = Ar1[0]; a1.y = Ar1[1];
    const float2* Wk = Wp + (size_t)((k0 >> 1) + kh) * DIM;  // one b64 per subtile
    const float2 w0p = Wk[col0];
    const float2 w1p = Wk[col0 + 16];
    b0.x = w0p.x; b0.y = w0p.y;
    b1.x = w1p.x; b1.y = w1p.y;
    c00 = wmma4(a0, b0, c00);
    c01 = wmma4(a0, b1, c01);
    c10 = wmma4(a1, b0, c10);
    c11 = wmma4(a1, b1, c11);
  }

  // C/D layout: lane L -> col = L&15 (per subtile), row = v + 8*(L>>4)
  const int   rb    = kh * 8;
  const float bias0 = bias[col0];
  const float bias1 = bias[col0 + 16];
#pragma unroll
  for (int v = 0; v < 8; ++v) {
    float x0 = c00[v] + bias0;
    float x1 = c01[v] + bias1;
    float x2 = c10[v] + bias0;
    float x3 = c11[v] + bias1;
    if (relu) {
      x0 = fmaxf(x0, 0.f); x1 = fmaxf(x1, 0.f);
      x2 = fmaxf(x2, 0.f); x3 = fmaxf(x3, 0.f);
    }
    dstH[(rb + v) * HS + col0]           = x0;
    dstH[(rb + v) * HS + col0 + 16]      = x1;
    dstH[(16 + rb + v) * HS + col0]      = x2;
    dstH[(16 + rb + v) * HS + col0 + 16] = x3;
  }
}

// LayerNorm stats over H[32][HS], all 128 threads participate.
__device__ __forceinline__ void ln_stats(const float* __restrict__ H,
                                         float (*part)[MT],
                                         float* __restrict__ rMu,
                                         float* __restrict__ rRs, int t) {
  const int r = t & 31, p = t >> 5;
  {
    const float* row = H + r * HS + p * 32;
    float s = 0.f;
#pragma unroll
    for (int c = 0; c < 32; ++c) s += row[c];
    part[p][r] = s;
  }
  __syncthreads();
  if (t < MT)
    rMu[t] = (part[0][t] + part[1][t] + part[2][t] + part[3][t]) * (1.f / DIM);
  __syncthreads();
  {
    const float  mu  = rMu[r];
    const float* row = H + r * HS + p * 32;
    float s = 0.f;
#pragma unroll
    for (int c = 0; c < 32; ++c) { float d = row[c] - mu; s += d * d; }
    part[p][r] = s;
  }
  __syncthreads();
  if (t < MT)
    rRs[t] = rsqrtf((part[0][t] + part[1][t] + part[2][t] + part[3][t]) *
                        (1.f / DIM) + LN_EPS);
  __syncthreads();
}

// ---------------- Edge update: MLP + LayerNorm + segment-sum (atomics) --------
__global__ __launch_bounds__(128) void edge_layer(
    const float* __restrict__ nodesW, float* __restrict__ edgesW,
    const int* __restrict__ nbr,
    const float2* __restrict__ w0, const float* __restrict__ b0,
    const float2* __restrict__ w1, const float* __restrict__ b1,
    const float2* __restrict__ w2, const float* __restrict__ b2,
    const float* __restrict__ lng, const float* __restrict__ lnb,
    float* __restrict__ agg) {
  __shared__ float Abuf[MT * AE_STRIDE];   // 32 x 384 gathered input
  __shared__ float H1[MT * HS];
  __shared__ float H2[MT * HS];
  __shared__ float part[4][MT];
  __shared__ float rMu[MT], rRs[MT];
  __shared__ int   recvS[MT];

  const int t = threadIdx.x;
  const int wave = t >> 5, lane = t & 31;
  const int e0 = blockIdx.x * MT;

  {  // Stage A: 4 threads per row, float4 loads of the 3 concat segments
    const int r = t >> 2, q = t & 3;
    const int e = e0 + r;
    const int recv = nbr[2 * e];
    const int send = nbr[2 * e + 1];
    if (q == 0) recvS[r] = recv;
    const float4* se = (const float4*)(edgesW + (size_t)e * DIM);
    const float4* sr = (const float4*)(nodesW + (size_t)recv * DIM);
    const float4* ss = (const float4*)(nodesW + (size_t)send * DIM);
    float4* d = (float4*)(Abuf + r * AE_STRIDE);
#pragma unroll
    for (int i = q; i < 32; i += 4) {
      d[i]      = se[i];
      d[i + 32] = sr[i];
      d[i + 64] = ss[i];
    }
  }
  __syncthreads();

  gemm_stage<KE, AE_STRIDE>(Abuf, w0, b0, H1, wave, lane, true);
  __syncthreads();
  gemm_stage<DIM, HS>(H1, w1, b1, H2, wave, lane, true);
  __syncthreads();
  gemm_stage<DIM, HS>(H2, w2, b2, H1, wave, lane, false);
  __syncthreads();

  ln_stats(H1, part, rMu, rRs, t);

  const float g = lng[t], bb = lnb[t];
  for (int r = 0; r < MT; ++r) {
    const float val = (H1[r * HS + t] - rMu[r]) * rRs[r] * g + bb;
    edgesW[(size_t)(e0 + r) * DIM + t] = val;
    atomicAdd(&agg[(size_t)recvS[r] * DIM + t], val);  // segment_sum
  }
}

// ---------------- Node update: MLP + LayerNorm (in-place on nodesW) -----------
__global__ __launch_bounds__(128) void node_layer(
    float* __restrict__ nodesW, const float* __restrict__ agg,
    const float2* __restrict__ w0, const float* __restrict__ b0,
    const float2* __restrict__ w1, const float* __restrict__ b1,
    const float2* __restrict__ w2, const float* __restrict__ b2,
    const float* __restrict__ lng, const float* __restrict__ lnb) {
  __shared__ float Abuf[MT * AN_STRIDE];
  __shared__ float H1[MT * HS];
  __shared__ float H2[MT * HS];
  __shared__ float part[4][MT];
  __shared__ float rMu[MT], rRs[MT];

  const int t = threadIdx.x;
  const int wave = t >> 5, lane = t & 31;
  const int n0 = blockIdx.x * MT;

  {  // Stage A = [agg | nodes], clamp tail rows (read-before-write, same block)
    const int r = t >> 2, q = t & 3;
    int n = n0 + r;
    if (n >= N_NODES) n = N_NODES - 1;
    const float4* sa = (const float4*)(agg + (size_t)n * DIM);
    const float4* sn = (const float4*)(nodesW + (size_t)n * DIM);
    float4* d = (float4*)(Abuf + r * AN_STRIDE);
#pragma unroll
    for (int i = q; i < 32; i += 4) {
      d[i]      = sa[i];
      d[i + 32] = sn[i];
    }
  }
  __syncthreads();

  gemm_stage<KN, AN_STRIDE>(Abuf, w0, b0, H1, wave, lane, true);
  __syncthreads();
  gemm_stage<DIM, HS>(H1, w1, b1, H2, wave, lane, true);
  __syncthreads();
  gemm_stage<DIM, HS>(H2, w2, b2, H1, wave, lane, false);
  __syncthreads();

  ln_stats(H1, part, rMu, rRs, t);

  const float g = lng[t], bb = lnb[t];
  for (int r = 0; r < MT; ++r) {
    const int n = n0 + r;
    if (n < N_NODES)
      nodesW[(size_t)n * DIM + t] = (H1[r * HS + t] - rMu[r]) * rRs[r] * g + bb;
  }
}

// ---------------- util -------------------------------------------------------
__global__ void copy_f4(const float4* __restrict__ s, float4* __restrict__ d, int n) {
  int i = blockIdx.x * blockDim.x + threadIdx.x;
  if (i < n) d[i] = s[i];
}
__global__ void zero_f4(float4* __restrict__ d, int n) {
  int i = blockIdx.x * blockDim.x + threadIdx.x;
  if (i < n) d[i] = make_float4(0.f, 0.f, 0.f, 0.f);
}
// Repack weight rows: dst[(k>>1)*256 + c*2 + (k&1)] = src[k*128 + c]
__global__ void pack_pairs(const float* __restrict__ src, float* __restrict__ dst,
                           int n) {
  int i = blockIdx.x * blockDim.x + threadIdx.x;
  if (i >= n) return;
  const int k = i >> 7, c = i & 127;
  dst[((k >> 1) << 8) + (c << 1) + (k & 1)] = src[i];
}

extern "C" void kernel_launch(void* const* d_in, const int* in_sizes, int n_in,
                              void* d_out, int out_size, void* d_ws, size_t ws_size,
                              hipStream_t stream) {
  const float* nodes_in = (const float*)d_in[0];
  const float* edges_in = (const float*)d_in[1];
  const int*   nbr      = (const int*)d_in[2];
  const float* ew0 = (const float*)d_in[3];
  const float* eb0 = (const float*)d_in[4];
  const float* ew1 = (const float*)d_in[5];
  const float* eb1 = (const float*)d_in[6];
  const float* ew2 = (const float*)d_in[7];
  const float* eb2 = (const float*)d_in[8];
  const float* elg = (const float*)d_in[9];
  const float* elb = (const float*)d_in[10];
  const float* nw0 = (const float*)d_in[11];
  const float* nb0 = (const float*)d_in[12];
  const float* nw1 = (const float*)d_in[13];
  const float* nb1 = (const float*)d_in[14];
  const float* nw2 = (const float*)d_in[15];
  const float* nb2 = (const float*)d_in[16];
  const float* nlg = (const float*)d_in[17];
  const float* nlb = (const float*)d_in[18];

  // d_out doubles as working storage: [nodes | edges | neighbor_idxs]
  float* out    = (float*)d_out;
  float* nodesW = out;
  float* edgesW = out + (size_t)N_NODES * DIM;
  float* idxOut = edgesW + (size_t)N_EDGES * DIM;

  // d_ws layout (floats): agg, then 6 pair-packed weight tensors (all layers)
  float* agg  = (float*)d_ws;                    // 3,840,000 floats
  float* pEw0 = agg  + (size_t)N_NODES * DIM;    // 10*384*128
  float* pEw1 = pEw0 + (size_t)N_LAYERS * KE * DIM;
  float* pEw2 = pEw1 + (size_t)N_LAYERS * DIM * DIM;
  float* pNw0 = pEw2 + (size_t)N_LAYERS * DIM * DIM;
  float* pNw1 = pNw0 + (size_t)N_LAYERS * KN * DIM;
  float* pNw2 = pNw1 + (size_t)N_LAYERS * DIM * DIM;

  const int nNode4 = N_NODES * DIM / 4;  // 960000
  const int nEdge4 = N_EDGES * DIM / 4;  // 7680000
  const int nIdx4  = N_EDGES * 2 / 4;    // 120000

  copy_f4<<<(nNode4 + 255) / 256, 256, 0, stream>>>((const float4*)nodes_in,
                                                    (float4*)nodesW, nNode4);
  copy_f4<<<(nEdge4 + 255) / 256, 256, 0, stream>>>((const float4*)edges_in,
                                                    (float4*)edgesW, nEdge4);
  copy_f4<<<(nIdx4 + 255) / 256, 256, 0, stream>>>((const float4*)nbr,
                                                   (float4*)idxOut, nIdx4);

  {  // pair-pack all weights (K even everywhere, layer slices stay contiguous)
    const int nE0 = N_LAYERS * KE * DIM, nEH = N_LAYERS * DIM * DIM;
    const int nN0 = N_LAYERS * KN * DIM;
    pack_pairs<<<(nE0 + 255) / 256, 256, 0, stream>>>(ew0, pEw0, nE0);
    pack_pairs<<<(nEH + 255) / 256, 256, 0, stream>>>(ew1, pEw1, nEH);
    pack_pairs<<<(nEH + 255) / 256, 256, 0, stream>>>(ew2, pEw2, nEH);
    pack_pairs<<<(nN0 + 255) / 256, 256, 0, stream>>>(nw0, pNw0, nN0);
    pack_pairs<<<(nEH + 255) / 256, 256, 0, stream>>>(nw1, pNw1, nEH);
    pack_pairs<<<(nEH + 255) / 256, 256, 0, stream>>>(nw2, pNw2, nEH);
  }

  const int edgeBlocks = N_EDGES / MT;             // 7500
  const int nodeBlocks = (N_NODES + MT - 1) / MT;  // 938

  for (int l = 0; l < N_LAYERS; ++l) {
    zero_f4<<<(nNode4 + 255) / 256, 256, 0, stream>>>((float4*)agg, nNode4);
    edge_layer<<<edgeBlocks, 128, 0, stream>>>(
        nodesW, edgesW, nbr,
        (const float2*)(pEw0 + (size_t)l * KE * DIM),  eb0 + (size_t)l * DIM,
        (const float2*)(pEw1 + (size_t)l * DIM * DIM), eb1 + (size_t)l * DIM,
        (const float2*)(pEw2 + (size_t)l * DIM * DIM), eb2 + (size_t)l * DIM,
        elg + (size_t)l * DIM, elb + (size_t)l * DIM, agg);
    node_layer<<<nodeBlocks, 128, 0, stream>>>(
        nodesW, agg,
        (const float2*)(pNw0 + (size_t)l * KN * DIM),  nb0 + (size_t)l * DIM,
        (const float2*)(pNw1 + (size_t)l * DIM * DIM), nb1 + (size_t)l * DIM,
        (const float2*)(pNw2 + (size_t)l * DIM * DIM), nb2 + (size_t)l * DIM,
        nlg + (size_t)l * DIM, nlb + (size_t)l * DIM);
  }
}